// ACCSLP_59485297050024
// MI455X (gfx1250) — compile-verified
//
#include <hip/hip_runtime.h>
#include <math.h>

// ---------------------------------------------------------------------------
// Tri-factor NMF multiplicative updates on MI455X (gfx1250, wave32).
// All dense GEMMs use V_WMMA_F32_16X16X4_F32 (fp32 matrix pipe) to match the
// fp32 reference numerics. Predictions are fused with the element-wise ratio
// (Div = M / (A@B)) and with the final log-objective so the N*N "pred"
// matrices are never materialized.
// ---------------------------------------------------------------------------

typedef __attribute__((ext_vector_type(2))) float v2f;
typedef __attribute__((ext_vector_type(8))) float v8f;

constexpr int   NDIM   = 4096;
constexpr int   RANK   = 128;
constexpr float ALPHAc = 0.5f;
constexpr float BETAc  = 0.5f;
constexpr int   NITERS = 2;
constexpr int   CPARTS = 16;   // column-sum partial phases (deterministic)

// D = A(16x4 f32) x B(4x16 f32) + C  via fp32 WMMA.
// Assumed per-ISA layout: A: lanes 0-15 row=lane, {K0,K1}; lanes 16-31 {K2,K3}.
//                         B: lanes 0-15 col=lane, {K0,K1}; lanes 16-31 {K2,K3}.
//                         C/D: VGPR g, lane t -> (row = g + 8*(t>>4), col = t&15).
__device__ inline v8f wmma4(v2f a, v2f b, v8f c) {
  return __builtin_amdgcn_wmma_f32_16x16x4_f32(
      /*neg_a=*/false, a, /*neg_b=*/false, b,
      /*c_mod=*/(short)0, c, /*reuse_a=*/false, /*reuse_b=*/false);
}

__device__ inline float wredf(float s) {
  for (int o = 16; o; o >>= 1) s += __shfl_down(s, o, 32);
  return s;
}
__device__ inline double wredd(double s) {
  for (int o = 16; o; o >>= 1) s += __shfl_down(s, o, 32);
  return s;
}

// Dv = M / (Af(N x R) @ Bf(R x N)).  One 16x16 output tile per wave.
// grid = (N/128, N/16), block = 256 (8 waves).
__global__ void k_predict_div(const float* __restrict__ M,
                              const float* __restrict__ Af,
                              const float* __restrict__ Bf,
                              float* __restrict__ Dv) {
  const int t    = threadIdx.x & 31;
  const int w    = threadIdx.x >> 5;
  const int l16  = t & 15;
  const int half = t >> 4;
  const int r0   = blockIdx.y << 4;
  const int c0   = ((blockIdx.x << 3) + w) << 4;

  v8f acc = {};
  const float* arow = Af + (size_t)(r0 + l16) * RANK + (half << 1);
  for (int k0 = 0; k0 < RANK; k0 += 4) {
    v2f a = *(const v2f*)(arow + k0);
    const int kA = k0 + (half << 1);
    v2f b;
    b.x = Bf[(size_t)kA * NDIM + c0 + l16];
    b.y = Bf[(size_t)(kA + 1) * NDIM + c0 + l16];
    acc = wmma4(a, b, acc);
  }
#pragma unroll
  for (int g = 0; g < 8; ++g) {
    const int m = g + (half << 3);
    const size_t idx = (size_t)(r0 + m) * NDIM + c0 + l16;
    Dv[idx] = M[idx] / acc[g];
  }
}

// accum += weight * sum( pred - M*log(pred) ),  pred = Af @ Bf.
__global__ void k_objective(const float* __restrict__ M,
                            const float* __restrict__ Af,
                            const float* __restrict__ Bf,
                            double* __restrict__ accum, float weight) {
  const int t    = threadIdx.x & 31;
  const int w    = threadIdx.x >> 5;
  const int l16  = t & 15;
  const int half = t >> 4;
  const int r0   = blockIdx.y << 4;
  const int c0   = ((blockIdx.x << 3) + w) << 4;

  v8f acc = {};
  const float* arow = Af + (size_t)(r0 + l16) * RANK + (half << 1);
  for (int k0 = 0; k0 < RANK; k0 += 4) {
    v2f a = *(const v2f*)(arow + k0);
    const int kA = k0 + (half << 1);
    v2f b;
    b.x = Bf[(size_t)kA * NDIM + c0 + l16];
    b.y = Bf[(size_t)(kA + 1) * NDIM + c0 + l16];
    acc = wmma4(a, b, acc);
  }
  double s = 0.0;
#pragma unroll
  for (int g = 0; g < 8; ++g) {
    const int m = g + (half << 3);
    const size_t idx = (size_t)(r0 + m) * NDIM + c0 + l16;
    const float p = acc[g];
    s += (double)(p - M[idx] * logf(p));
  }
  s = wredd(s);
  if (t == 0) atomicAdd(accum, (double)weight * s);
}

// U *= (Sdiv @ Hw1^T + Zdiv @ Hw2^T).  Output N x R; wave w -> rank tile.
// grid = N/16, block = 256 (8 waves cover RANK/16 = 8 column tiles).
__global__ void k_update_U(float* __restrict__ U,
                           const float* __restrict__ Sdiv,
                           const float* __restrict__ Zdiv,
                           const float* __restrict__ Hw1,
                           const float* __restrict__ Hw2) {
  const int t    = threadIdx.x & 31;
  const int w    = threadIdx.x >> 5;
  const int l16  = t & 15;
  const int half = t >> 4;
  const int r0   = blockIdx.x << 4;   // output row (l)
  const int c0   = w << 4;            // output col (rank k)

  v8f acc1 = {}, acc2 = {};
  const float* a1row = Sdiv + (size_t)(r0 + l16) * NDIM + (half << 1);
  const float* a2row = Zdiv + (size_t)(r0 + l16) * NDIM + (half << 1);
  const float* b1row = Hw1  + (size_t)(c0 + l16) * NDIM + (half << 1);
  const float* b2row = Hw2  + (size_t)(c0 + l16) * NDIM + (half << 1);
  for (int j0 = 0; j0 < NDIM; j0 += 4) {
    v2f a1 = *(const v2f*)(a1row + j0);
    v2f b1 = *(const v2f*)(b1row + j0);
    v2f a2 = *(const v2f*)(a2row + j0);
    v2f b2 = *(const v2f*)(b2row + j0);
    acc1 = wmma4(a1, b1, acc1);
    acc2 = wmma4(a2, b2, acc2);
  }
#pragma unroll
  for (int g = 0; g < 8; ++g) {
    const int m = g + (half << 3);
    const size_t idx = (size_t)(r0 + m) * RANK + c0 + l16;
    U[idx] *= (acc1[g] + acc2[g]);
  }
}

// H *= (Uw1^T @ Sdiv + Uw2^T @ Xdiv).  Output R x N; wave w -> rank-row tile.
// grid = N/16, block = 256.
__global__ void k_update_H(float* __restrict__ H,
                           const float* __restrict__ Sdiv,
                           const float* __restrict__ Xdiv,
                           const float* __restrict__ Uw1,
                           const float* __restrict__ Uw2) {
  const int t    = threadIdx.x & 31;
  const int w    = threadIdx.x >> 5;
  const int l16  = t & 15;
  const int half = t >> 4;
  const int r0   = w << 4;            // output row (rank k)
  const int c0   = blockIdx.x << 4;   // output col (j)

  v8f acc1 = {}, acc2 = {};
  for (int l0 = 0; l0 < NDIM; l0 += 4) {
    const int lA = l0 + (half << 1);
    v2f a1, a2, b1, b2;
    a1.x = Uw1[(size_t)lA * RANK + r0 + l16];
    a1.y = Uw1[(size_t)(lA + 1) * RANK + r0 + l16];
    b1.x = Sdiv[(size_t)lA * NDIM + c0 + l16];
    b1.y = Sdiv[(size_t)(lA + 1) * NDIM + c0 + l16];
    a2.x = Uw2[(size_t)lA * RANK + r0 + l16];
    a2.y = Uw2[(size_t)(lA + 1) * RANK + r0 + l16];
    b2.x = Xdiv[(size_t)lA * NDIM + c0 + l16];
    b2.y = Xdiv[(size_t)(lA + 1) * NDIM + c0 + l16];
    acc1 = wmma4(a1, b1, acc1);
    acc2 = wmma4(a2, b2, acc2);
  }
#pragma unroll
  for (int g = 0; g < 8; ++g) {
    const int m = g + (half << 3);
    const size_t idx = (size_t)(r0 + m) * NDIM + c0 + l16;
    H[idx] *= (acc1[g] + acc2[g]);
  }
}

// O1 = P/(P+sc*Q); O2 = sc*Q/(P+sc*Q).  (N*R elements)
__global__ void k_prep2(const float* __restrict__ P, const float* __restrict__ Q,
                        float sc, float* __restrict__ O1, float* __restrict__ O2) {
  const size_t i = (size_t)blockIdx.x * blockDim.x + threadIdx.x;
  const float p = P[i], q = sc * Q[i];
  const float inv = 1.0f / (p + q);
  O1[i] = p * inv;
  O2[i] = q * inv;
}

// rs[row] = sum_j Dv[row, j].  One wave per row; grid = N/8, block = 256.
__global__ void k_rowsum(const float* __restrict__ Dv, float* __restrict__ rs) {
  const int t = threadIdx.x & 31;
  const int w = threadIdx.x >> 5;
  const int row = (blockIdx.x << 3) + w;
  float s = 0.0f;
  const float* p = Dv + (size_t)row * NDIM;
  for (int j = t; j < NDIM; j += 32) s += p[j];
  s = wredf(s);
  if (t == 0) rs[row] = s;
}

// W[l,k] *= rs[l]  (row-major N x R)
__global__ void k_scale_rows(float* __restrict__ W, const float* __restrict__ rs) {
  const size_t i = (size_t)blockIdx.x * blockDim.x + threadIdx.x;
  W[i] *= rs[i >> 7];  // i / RANK
}

// Phase 1 of deterministic column sum: part[by, col] = sum of 256 rows.
__global__ void k_colpart(const float* __restrict__ Dv, float* __restrict__ part) {
  const int col = blockIdx.x * 256 + threadIdx.x;
  const int rb  = blockIdx.y * (NDIM / CPARTS);
  float s = 0.0f;
  for (int r = 0; r < NDIM / CPARTS; ++r)
    s += Dv[(size_t)(rb + r) * NDIM + col];
  part[(size_t)blockIdx.y * NDIM + col] = s;
}

// Phase 2: cs[col] = sum_p part[p, col]  (fixed order -> deterministic)
__global__ void k_colreduce(const float* __restrict__ part, float* __restrict__ cs) {
  const int col = blockIdx.x * 256 + threadIdx.x;
  float s = 0.0f;
#pragma unroll
  for (int p = 0; p < CPARTS; ++p) s += part[(size_t)p * NDIM + col];
  cs[col] = s;
}

// V[k,j] *= cs[j]  (row-major R x N)
__global__ void k_scale_cols(float* __restrict__ V, const float* __restrict__ cs) {
  const size_t i = (size_t)blockIdx.x * blockDim.x + threadIdx.x;
  V[i] *= cs[i & (NDIM - 1)];
}

__global__ void k_finalize(const double* __restrict__ a, float* __restrict__ out) {
  if (threadIdx.x == 0 && blockIdx.x == 0) out[0] = (float)a[0];
}

extern "C" void kernel_launch(void* const* d_in, const int* in_sizes, int n_in,
                              void* d_out, int out_size, void* d_ws, size_t ws_size,
                              hipStream_t stream) {
  (void)in_sizes; (void)n_in; (void)out_size; (void)ws_size;
  const float* S  = (const float*)d_in[0];
  const float* Z  = (const float*)d_in[1];
  const float* X  = (const float*)d_in[2];
  const float* U0 = (const float*)d_in[3];
  const float* H0 = (const float*)d_in[4];
  const float* W0 = (const float*)d_in[5];
  const float* V0 = (const float*)d_in[6];

  const size_t NN = (size_t)NDIM * NDIM;
  const size_t NR = (size_t)NDIM * RANK;

  char* p = (char*)d_ws;
  float* Sdiv = (float*)p; p += NN * 4;
  float* Xdiv = (float*)p; p += NN * 4;
  float* Zdiv = (float*)p; p += NN * 4;
  float* Uc   = (float*)p; p += NR * 4;
  float* Hc   = (float*)p; p += NR * 4;
  float* Wc   = (float*)p; p += NR * 4;
  float* Vc   = (float*)p; p += NR * 4;
  float* Hw1  = (float*)p; p += NR * 4;
  float* Hw2  = (float*)p; p += NR * 4;
  float* Uw1  = (float*)p; p += NR * 4;
  float* Uw2  = (float*)p; p += NR * 4;
  float* rs   = (float*)p; p += (size_t)NDIM * 4;
  float* cs   = (float*)p; p += (size_t)NDIM * 4;
  float* cpart= (float*)p; p += (size_t)CPARTS * NDIM * 4;
  double* accd= (double*)p;

  // Work on private copies; inputs must stay untouched.
  hipMemcpyAsync(Uc, U0, NR * 4, hipMemcpyDeviceToDevice, stream);
  hipMemcpyAsync(Hc, H0, NR * 4, hipMemcpyDeviceToDevice, stream);
  hipMemcpyAsync(Wc, W0, NR * 4, hipMemcpyDeviceToDevice, stream);
  hipMemcpyAsync(Vc, V0, NR * 4, hipMemcpyDeviceToDevice, stream);

  const dim3 gPD(NDIM / 128, NDIM / 16);
  const int ew = (int)(NR / 256);  // 2048 blocks for N*R element-wise kernels

  // Initial ratios.
  k_predict_div<<<gPD, 256, 0, stream>>>(S, Uc, Hc, Sdiv);
  k_predict_div<<<gPD, 256, 0, stream>>>(X, Wc, Hc, Xdiv);
  k_predict_div<<<gPD, 256, 0, stream>>>(Z, Uc, Vc, Zdiv);

  for (int it = 0; it < NITERS; ++it) {
    const bool last = (it == NITERS - 1);
    // --- U update
    k_prep2<<<ew, 256, 0, stream>>>(Hc, Vc, BETAc, Hw1, Hw2);
    k_update_U<<<NDIM / 16, 256, 0, stream>>>(Uc, Sdiv, Zdiv, Hw1, Hw2);
    k_predict_div<<<gPD, 256, 0, stream>>>(S, Uc, Hc, Sdiv);
    k_predict_div<<<gPD, 256, 0, stream>>>(Z, Uc, Vc, Zdiv);
    // --- H update (uses Xdiv from before this iteration)
    k_prep2<<<ew, 256, 0, stream>>>(Uc, Wc, ALPHAc, Uw1, Uw2);
    k_update_H<<<NDIM / 16, 256, 0, stream>>>(Hc, Sdiv, Xdiv, Uw1, Uw2);
    if (!last) k_predict_div<<<gPD, 256, 0, stream>>>(S, Uc, Hc, Sdiv);
    k_predict_div<<<gPD, 256, 0, stream>>>(X, Wc, Hc, Xdiv);
    // --- W update: W *= rowsum(Xdiv)
    k_rowsum<<<NDIM / 8, 256, 0, stream>>>(Xdiv, rs);
    k_scale_rows<<<ew, 256, 0, stream>>>(Wc, rs);
    if (!last) k_predict_div<<<gPD, 256, 0, stream>>>(X, Wc, Hc, Xdiv);
    // --- V update: V *= colsum(Zdiv)  (Zdiv from after the U update)
    k_colpart<<<dim3(NDIM / 256, CPARTS), 256, 0, stream>>>(Zdiv, cpart);
    k_colreduce<<<NDIM / 256, 256, 0, stream>>>(cpart, cs);
    k_scale_cols<<<ew, 256, 0, stream>>>(Vc, cs);
    if (!last) k_predict_div<<<gPD, 256, 0, stream>>>(Z, Uc, Vc, Zdiv);
  }

  // Final objective with final factors (fused predict + log terms).
  hipMemsetAsync(accd, 0, sizeof(double), stream);
  k_objective<<<gPD, 256, 0, stream>>>(S, Uc, Hc, accd, 1.0f);
  k_objective<<<gPD, 256, 0, stream>>>(X, Wc, Hc, accd, ALPHAc);
  k_objective<<<gPD, 256, 0, stream>>>(Z, Uc, Vc, accd, BETAc);
  k_finalize<<<1, 1, 0, stream>>>(accd, (float*)d_out);
}